// GAT_66683662238424
// MI455X (gfx1250) — compile-verified
//
#include <hip/hip_runtime.h>
#include <math.h>

// ---------------------------------------------------------------------------
// GAT (2-layer) for MI455X / gfx1250, wave32 + WMMA fp32.
//   GEMM1: x[N,256]@W1[256,64]  via v_wmma_f32_16x16x4_f32 (fp32-exact)
//   GEMM2: h[N,64]@W2[64,16]    via v_wmma_f32_16x16x4_f32
//   Edge softmax/scatter via L2-resident atomics (all node arrays << 192MB L2)
// ---------------------------------------------------------------------------

typedef float v2f __attribute__((ext_vector_type(2)));
typedef float v8f __attribute__((ext_vector_type(8)));

#define DIM_IN 256
#define HID    64
#define HEADS  8
#define DH     8
#define DOUT   16
#define NEG_SLOPE 0.2f
#define EPS_F 1e-16f
#define ENC_NEGINF 0x007FFFFFu  // enc(-inf)

// order-preserving float<->uint map so segment-max can use uint atomicMax
__device__ __forceinline__ unsigned enc_f32(float f) {
  unsigned u = __float_as_uint(f);
  return (u & 0x80000000u) ? ~u : (u | 0x80000000u);
}
__device__ __forceinline__ float dec_f32(unsigned e) {
  unsigned u = (e & 0x80000000u) ? (e ^ 0x80000000u) : ~e;
  return __uint_as_float(u);
}
__device__ __forceinline__ float leaky(float v) {
  return v > 0.0f ? v : NEG_SLOPE * v;
}

// ---------------- GEMM1: h1 = x @ W1 ---------------------------------------
// 256 threads = 8 waves. Each wave owns one 16x16 output tile:
//   waves 0..3 -> row-tile 2*bid+0, col tiles 0..3
//   waves 4..7 -> row-tile 2*bid+1, col tiles 0..3
__global__ __launch_bounds__(256) void gat_gemm1_wmma(
    const float* __restrict__ x, const float* __restrict__ W1,
    float* __restrict__ h1, int N) {
  __shared__ float sW[DIM_IN * HID];  // 64KB: whole W1 in LDS
  int tid = threadIdx.x;
  {
    const float4* Wv = (const float4*)W1;
    float4* sWv = (float4*)sW;
#pragma unroll
    for (int i = 0; i < 16; ++i) sWv[tid + i * 256] = Wv[tid + i * 256];
  }
  __syncthreads();

  int wave = tid >> 5, lane = tid & 31;
  int row0 = (blockIdx.x * 2 + (wave >> 2)) * 16;
  int col0 = (wave & 3) * 16;
  if (row0 >= N) return;

  int r  = lane & 15;   // A-row within tile == B/C column within tile
  int kh = lane >> 4;   // which K-half this lane holds

  int arow_idx = row0 + r; if (arow_idx >= N) arow_idx = N - 1;
  const float* arow  = x + (size_t)arow_idx * DIM_IN + 2 * kh;
  const float* bbase = sW + col0 + r;

  v8f c = {};
  for (int k = 0; k < DIM_IN; k += 4) {
    if ((k & 63) == 0) __builtin_prefetch(arow + k + 64, 0, 3);  // WGP-scope prefetch
    float2 av = *(const float2*)(arow + k);
    v2f a; a.x = av.x; a.y = av.y;
    v2f b; b.x = bbase[(k + 2 * kh) * HID];
           b.y = bbase[(k + 2 * kh + 1) * HID];
    c = __builtin_amdgcn_wmma_f32_16x16x4_f32(false, a, false, b,
                                              (short)0, c, false, false);
  }
  float* orow = h1 + (size_t)row0 * HID + col0 + r;
  if (row0 + 16 <= N) {  // full tile fast path: unguarded store clause
#pragma unroll
    for (int rr = 0; rr < 8; ++rr)
      orow[(size_t)(rr + 8 * kh) * HID] = c[rr];
  } else {
#pragma unroll
    for (int rr = 0; rr < 8; ++rr) {
      int row = row0 + rr + 8 * kh;
      if (row < N) orow[(size_t)(rr + 8 * kh) * HID] = c[rr];
    }
  }
}

// ---------------- GEMM2: h2 = hact @ W2 ------------------------------------
__global__ __launch_bounds__(256) void gat_gemm2_wmma(
    const float* __restrict__ hact, const float* __restrict__ W2,
    float* __restrict__ h2, int N) {
  __shared__ float sW[HID * DOUT];  // 4KB
  int tid = threadIdx.x;
  {
    const float4* Wv = (const float4*)W2;
    float4* sWv = (float4*)sW;
    sWv[tid] = Wv[tid];  // 1024 floats = 256 float4
  }
  __syncthreads();

  int wave = tid >> 5, lane = tid & 31;
  int row0 = (blockIdx.x * 8 + wave) * 16;
  if (row0 >= N) return;

  int r  = lane & 15;
  int kh = lane >> 4;
  int arow_idx = row0 + r; if (arow_idx >= N) arow_idx = N - 1;
  const float* arow  = hact + (size_t)arow_idx * HID + 2 * kh;
  const float* bbase = sW + r;

  v8f c = {};
#pragma unroll
  for (int k = 0; k < HID; k += 4) {
    float2 av = *(const float2*)(arow + k);
    v2f a; a.x = av.x; a.y = av.y;
    v2f b; b.x = bbase[(k + 2 * kh) * DOUT];
           b.y = bbase[(k + 2 * kh + 1) * DOUT];
    c = __builtin_amdgcn_wmma_f32_16x16x4_f32(false, a, false, b,
                                              (short)0, c, false, false);
  }
  float* orow = h2 + (size_t)row0 * DOUT + r;
  if (row0 + 16 <= N) {
#pragma unroll
    for (int rr = 0; rr < 8; ++rr)
      orow[(size_t)(rr + 8 * kh) * DOUT] = c[rr];
  } else {
#pragma unroll
    for (int rr = 0; rr < 8; ++rr) {
      int row = row0 + rr + 8 * kh;
      if (row < N) orow[(size_t)(rr + 8 * kh) * DOUT] = c[rr];
    }
  }
}

// ---------------- misc node kernels ----------------------------------------
__global__ void gat_fill_u32(unsigned* __restrict__ p, unsigned v, int n) {
  int t = blockIdx.x * blockDim.x + threadIdx.x;
  if (t < n) p[t] = v;
}

__global__ void gat_node_alpha1(const float* __restrict__ h1,
                                const float* __restrict__ a_src,
                                const float* __restrict__ a_dst,
                                float* __restrict__ as, float* __restrict__ ad,
                                int NH) {
  int t = blockIdx.x * blockDim.x + threadIdx.x;  // per (n, head)
  if (t >= NH) return;
  int n = t >> 3, h = t & 7;
  const float* hp = h1 + (size_t)n * HID + h * DH;
  const float* sp = a_src + h * DH;
  const float* dp = a_dst + h * DH;
  float ss = 0.f, dd = 0.f;
#pragma unroll
  for (int i = 0; i < DH; ++i) { float v = hp[i]; ss += v * sp[i]; dd += v * dp[i]; }
  as[t] = ss; ad[t] = dd;
}

__global__ void gat_node_alpha2(const float* __restrict__ h2,
                                const float* __restrict__ a_src,
                                const float* __restrict__ a_dst,
                                float* __restrict__ as, float* __restrict__ ad,
                                int N) {
  int n = blockIdx.x * blockDim.x + threadIdx.x;
  if (n >= N) return;
  const float* hp = h2 + (size_t)n * DOUT;
  float ss = 0.f, dd = 0.f;
#pragma unroll
  for (int i = 0; i < DOUT; ++i) { float v = hp[i]; ss += v * a_src[i]; dd += v * a_dst[i]; }
  as[n] = ss; ad[n] = dd;
}

// ---------------- edge passes, layer 1 (8 heads) ---------------------------
// alpha arrays are [N,8] = 32B/node, 16B aligned -> gather with b128 loads.
__global__ void gat_edge_max1(const int* __restrict__ src, const int* __restrict__ dst,
                              const float* __restrict__ as, const float* __restrict__ ad,
                              unsigned* __restrict__ m, int E) {
  int e = blockIdx.x * blockDim.x + threadIdx.x;
  if (e >= E) return;
  int s = src[e], d = dst[e];
  const float4* as4 = (const float4*)as;
  const float4* ad4 = (const float4*)ad;
  float4 sv[2] = { as4[(size_t)s * 2], as4[(size_t)s * 2 + 1] };
  float4 dv[2] = { ad4[(size_t)d * 2], ad4[(size_t)d * 2 + 1] };
  const float* sf = (const float*)sv;
  const float* df = (const float*)dv;
#pragma unroll
  for (int h = 0; h < HEADS; ++h) {
    float v = leaky(sf[h] + df[h]);
    atomicMax(&m[(size_t)d * HEADS + h], enc_f32(v));
  }
}

__global__ void gat_edge_sum1(const int* __restrict__ src, const int* __restrict__ dst,
                              const float* __restrict__ as, const float* __restrict__ ad,
                              const unsigned* __restrict__ m, float* __restrict__ ssum,
                              int E) {
  int e = blockIdx.x * blockDim.x + threadIdx.x;
  if (e >= E) return;
  int s = src[e], d = dst[e];
  const float4* as4 = (const float4*)as;
  const float4* ad4 = (const float4*)ad;
  const uint4*  m4  = (const uint4*)m;
  float4 sv[2] = { as4[(size_t)s * 2], as4[(size_t)s * 2 + 1] };
  float4 dv[2] = { ad4[(size_t)d * 2], ad4[(size_t)d * 2 + 1] };
  uint4  mv[2] = { m4[(size_t)d * 2],  m4[(size_t)d * 2 + 1] };
  const float*    sf = (const float*)sv;
  const float*    df = (const float*)dv;
  const unsigned* mu = (const unsigned*)mv;
#pragma unroll
  for (int h = 0; h < HEADS; ++h) {
    float v = leaky(sf[h] + df[h]);
    atomicAdd(&ssum[(size_t)d * HEADS + h], __expf(v - dec_f32(mu[h])));
  }
}

__global__ void gat_edge_scatter1(const int* __restrict__ src, const int* __restrict__ dst,
                                  const float* __restrict__ as, const float* __restrict__ ad,
                                  const unsigned* __restrict__ m, const float* __restrict__ ssum,
                                  const float* __restrict__ h1, float* __restrict__ out1,
                                  unsigned total) {
  unsigned t = blockIdx.x * blockDim.x + threadIdx.x;  // per (edge, component)
  if (t >= total) return;
  int e = (int)(t >> 6), c = (int)(t & 63), h = c >> 3;
  int s = src[e], d = dst[e];
  float v = leaky(as[(size_t)s * HEADS + h] + ad[(size_t)d * HEADS + h]);
  float ex = __expf(v - dec_f32(m[(size_t)d * HEADS + h]));
  float alpha = ex / (ssum[(size_t)d * HEADS + h] + EPS_F);
  atomicAdd(&out1[(size_t)d * HID + c], alpha * h1[(size_t)s * HID + c]);
}

__global__ void gat_elu_bias(float* __restrict__ out1, const float* __restrict__ b1,
                             int total) {
  int t = blockIdx.x * blockDim.x + threadIdx.x;
  if (t >= total) return;
  float v = out1[t] + b1[t & (HID - 1)];
  out1[t] = v > 0.0f ? v : expm1f(v);
}

// ---------------- edge passes, layer 2 (1 head, 16 dims) -------------------
__global__ void gat_edge_max2(const int* __restrict__ src, const int* __restrict__ dst,
                              const float* __restrict__ as, const float* __restrict__ ad,
                              unsigned* __restrict__ m, int E) {
  int e = blockIdx.x * blockDim.x + threadIdx.x;
  if (e >= E) return;
  int s = src[e], d = dst[e];
  float v = leaky(as[s] + ad[d]);
  atomicMax(&m[d], enc_f32(v));
}

__global__ void gat_edge_sum2(const int* __restrict__ src, const int* __restrict__ dst,
                              const float* __restrict__ as, const float* __restrict__ ad,
                              const unsigned* __restrict__ m, float* __restrict__ ssum,
                              int E) {
  int e = blockIdx.x * blockDim.x + threadIdx.x;
  if (e >= E) return;
  int s = src[e], d = dst[e];
  float v = leaky(as[s] + ad[d]);
  atomicAdd(&ssum[d], __expf(v - dec_f32(m[d])));
}

__global__ void gat_edge_scatter2(const int* __restrict__ src, const int* __restrict__ dst,
                                  const float* __restrict__ as, const float* __restrict__ ad,
                                  const unsigned* __restrict__ m, const float* __restrict__ ssum,
                                  const float* __restrict__ h2, float* __restrict__ out2,
                                  unsigned total) {
  unsigned t = blockIdx.x * blockDim.x + threadIdx.x;  // per (edge, component)
  if (t >= total) return;
  int e = (int)(t >> 4), c = (int)(t & 15);
  int s = src[e], d = dst[e];
  float v = leaky(as[s] + ad[d]);
  float ex = __expf(v - dec_f32(m[d]));
  float alpha = ex / (ssum[d] + EPS_F);
  atomicAdd(&out2[(size_t)d * DOUT + c], alpha * h2[(size_t)s * DOUT + c]);
}

// ---------------- finalize: bias + log_softmax -----------------------------
__global__ void gat_finalize(const float* __restrict__ out2, const float* __restrict__ b2,
                             float* __restrict__ dout, int N) {
  int n = blockIdx.x * blockDim.x + threadIdx.x;
  if (n >= N) return;
  float v[DOUT];
  float mx = -INFINITY;
#pragma unroll
  for (int c = 0; c < DOUT; ++c) {
    v[c] = out2[(size_t)n * DOUT + c] + b2[c];
    mx = fmaxf(mx, v[c]);
  }
  float se = 0.f;
#pragma unroll
  for (int c = 0; c < DOUT; ++c) se += __expf(v[c] - mx);
  float lse = mx + __logf(se);
#pragma unroll
  for (int c = 0; c < DOUT; ++c) {
    dout[(size_t)n * DOUT + c] = v[c];
    dout[(size_t)N * DOUT + (size_t)n * DOUT + c] = v[c] - lse;
  }
}

// ---------------------------------------------------------------------------
extern "C" void kernel_launch(void* const* d_in, const int* in_sizes, int n_in,
                              void* d_out, int out_size, void* d_ws, size_t ws_size,
                              hipStream_t stream) {
  const float* x   = (const float*)d_in[0];
  const int*   ei  = (const int*)d_in[1];
  const float* W1  = (const float*)d_in[2];
  const float* a1s = (const float*)d_in[3];
  const float* a1d = (const float*)d_in[4];
  const float* b1  = (const float*)d_in[5];
  const float* W2  = (const float*)d_in[6];
  const float* a2s = (const float*)d_in[7];
  const float* a2d = (const float*)d_in[8];
  const float* b2  = (const float*)d_in[9];

  int N = in_sizes[0] / DIM_IN;
  int E = in_sizes[1] / 2;
  const int* src = ei;
  const int* dst = ei + E;

  // workspace layout (region A reused by layer 2 after h1 dies)
  char* ws = (char*)d_ws;
  float*    h1   = (float*)ws;                              // N*64 (region A)
  float*    out1 = (float*)(ws + (size_t)N * HID * 4);      // N*64 (region B, becomes hact)
  char* p = ws + (size_t)N * HID * 4 * 2;
  float*    as1 = (float*)p;  p += (size_t)N * HEADS * 4;
  float*    ad1 = (float*)p;  p += (size_t)N * HEADS * 4;
  unsigned* m1  = (unsigned*)p; p += (size_t)N * HEADS * 4;
  float*    s1  = (float*)p;  p += (size_t)N * HEADS * 4;
  // layer-2 buffers overlap region A (h1), valid only after elu_bias
  float*    h2  = h1;                                       // N*16
  char* q = (char*)(h1 + (size_t)N * DOUT);
  float*    as2 = (float*)q;  q += (size_t)N * 4;
  float*    ad2 = (float*)q;  q += (size_t)N * 4;
  unsigned* m2  = (unsigned*)q; q += (size_t)N * 4;
  float*    s2  = (float*)q;  q += (size_t)N * 4;
  float*    out2= (float*)q;                                // N*16

  const int B = 256;
  // ---- layer 1 init ----
  hipMemsetAsync(s1, 0, (size_t)N * HEADS * 4, stream);
  hipMemsetAsync(out1, 0, (size_t)N * HID * 4, stream);
  gat_fill_u32<<<(N * HEADS + B - 1) / B, B, 0, stream>>>(m1, ENC_NEGINF, N * HEADS);

  // ---- layer 1 ----
  gat_gemm1_wmma<<<(N + 31) / 32, B, 0, stream>>>(x, W1, h1, N);
  gat_node_alpha1<<<(N * HEADS + B - 1) / B, B, 0, stream>>>(h1, a1s, a1d, as1, ad1, N * HEADS);
  gat_edge_max1<<<(E + B - 1) / B, B, 0, stream>>>(src, dst, as1, ad1, m1, E);
  gat_edge_sum1<<<(E + B - 1) / B, B, 0, stream>>>(src, dst, as1, ad1, m1, s1, E);
  {
    unsigned total = (unsigned)E * HID;
    gat_edge_scatter1<<<(total + B - 1) / B, B, 0, stream>>>(src, dst, as1, ad1, m1, s1,
                                                             h1, out1, total);
  }
  gat_elu_bias<<<(N * HID + B - 1) / B, B, 0, stream>>>(out1, b1, N * HID);

  // ---- layer 2 init (region A is free now) ----
  hipMemsetAsync(s2, 0, (size_t)N * 4, stream);
  hipMemsetAsync(out2, 0, (size_t)N * DOUT * 4, stream);
  gat_fill_u32<<<(N + B - 1) / B, B, 0, stream>>>(m2, ENC_NEGINF, N);

  // ---- layer 2 ----
  gat_gemm2_wmma<<<(N + 127) / 128, B, 0, stream>>>(out1, W2, h2, N);
  gat_node_alpha2<<<(N + B - 1) / B, B, 0, stream>>>(h2, a2s, a2d, as2, ad2, N);
  gat_edge_max2<<<(E + B - 1) / B, B, 0, stream>>>(src, dst, as2, ad2, m2, E);
  gat_edge_sum2<<<(E + B - 1) / B, B, 0, stream>>>(src, dst, as2, ad2, m2, s2, E);
  {
    unsigned total = (unsigned)E * DOUT;
    gat_edge_scatter2<<<(total + B - 1) / B, B, 0, stream>>>(src, dst, as2, ad2, m2, s2,
                                                             h2, out2, total);
  }
  gat_finalize<<<(N + B - 1) / B, B, 0, stream>>>(out2, b2, (float*)d_out, N);
}